// HomoGNN_65386582114575
// MI455X (gfx1250) — compile-verified
//
#include <hip/hip_runtime.h>
#include <hip/hip_bf16.h>

#define NDIM 128
#define WAVES_PER_BLOCK 4

typedef __attribute__((ext_vector_type(2))) float v2f;
typedef __attribute__((ext_vector_type(8))) float v8f;

#if __has_builtin(__builtin_amdgcn_wmma_f32_16x16x4_f32)
#define HAVE_WMMA_F32X4 1
#else
#define HAVE_WMMA_F32X4 0
#endif

// ---------------------------------------------------------------------------
// zero a float buffer (n must be a multiple of 4)
// ---------------------------------------------------------------------------
__global__ __launch_bounds__(256) void zero_f4(float4* __restrict__ p, int n4) {
    int i = blockIdx.x * blockDim.x + threadIdx.x;
    int stride = gridDim.x * blockDim.x;
    float4 z = make_float4(0.f, 0.f, 0.f, 0.f);
    for (; i < n4; i += stride) p[i] = z;
}

// ---------------------------------------------------------------------------
// in-degree: deg[dst[e]] += 1
// ---------------------------------------------------------------------------
__global__ __launch_bounds__(256) void degree_kernel(const long long* __restrict__ dst,
                                                     float* __restrict__ deg, int n_edges) {
    int i = blockIdx.x * blockDim.x + threadIdx.x;
    int stride = gridDim.x * blockDim.x;
    for (; i < n_edges; i += stride) {
        atomicAdd(&deg[(int)dst[i]], 1.0f);
    }
}

// ---------------------------------------------------------------------------
// inv[i] = 1 / max(deg[i], 1)   (in place)
// ---------------------------------------------------------------------------
__global__ __launch_bounds__(256) void invdeg_kernel(float* __restrict__ deg, int n) {
    int i = blockIdx.x * blockDim.x + threadIdx.x;
    if (i < n) deg[i] = 1.0f / fmaxf(deg[i], 1.0f);
}

// ---------------------------------------------------------------------------
// edge scatter: agg[dst] += feat[src]   (one wave32 per edge, 4 floats/lane)
// ---------------------------------------------------------------------------
__global__ __launch_bounds__(256) void scatter_kernel(const float* __restrict__ feat,
                                                      const long long* __restrict__ src,
                                                      const long long* __restrict__ dst,
                                                      float* __restrict__ agg, int n_edges) {
    const int lane = threadIdx.x & 31;
    int wave = (blockIdx.x * blockDim.x + threadIdx.x) >> 5;
    const int nwaves = (gridDim.x * blockDim.x) >> 5;
    for (int e = wave; e < n_edges; e += nwaves) {
        const int s = (int)src[e];
        const int d = (int)dst[e];
        const float4 v = ((const float4*)(feat + (long)s * NDIM))[lane];
        float* ap = agg + (long)d * NDIM + lane * 4;
        atomicAdd(ap + 0, v.x);
        atomicAdd(ap + 1, v.y);
        atomicAdd(ap + 2, v.z);
        atomicAdd(ap + 3, v.w);
    }
}

// ---------------------------------------------------------------------------
// Fused SAGE layer:  out = act( (agg .* inv) @ Wl + h @ Wr + b )
// One wave per 16x16 output tile, fp32 WMMA 16x16x4.
//   tiles = (N/16) * (D/16) = 1250*8 = 10000 ; grid = 2500 blocks * 4 waves
// Fragment layouts per CDNA5 ISA 7.12.2:
//   A 16x4 : lane -> M = lane&15 ; VGPR pair holds K = k0 + 2*(lane>>4) {+0,+1}
//   B 4x16 : lane -> N = lane&15 ; VGPR pair holds K = k0 + 2*(lane>>4) {+0,+1}
//   C 16x16: VGPR v -> M = v + 8*(lane>>4), N = lane&15
// ---------------------------------------------------------------------------
__global__ __launch_bounds__(WAVES_PER_BLOCK * 32) void sage_layer_kernel(
    const float* __restrict__ agg, const float* __restrict__ inv,
    const float* __restrict__ h, const float* __restrict__ Wl,
    const float* __restrict__ Wr, const float* __restrict__ bias,
    float* __restrict__ out, int do_relu) {
    const int lane = threadIdx.x & 31;
    const int wave = threadIdx.x >> 5;
    const int tile = blockIdx.x * WAVES_PER_BLOCK + wave;
    const int M0 = (tile >> 3) * 16;   // node-row tile
    const int N0 = (tile & 7) * 16;    // output-feature tile

    const int mn    = lane & 15;       // M for A-frag, N for B/C-frag
    const int khalf = (lane >> 4) * 2; // 0 or 2: which K-pair this half-wave holds
    const int hi    = (lane >> 4) * 8; // C-row offset for upper half-wave

    const float* __restrict__ arow = agg + (long)(M0 + mn) * NDIM;
    const float* __restrict__ hrow = h + (long)(M0 + mn) * NDIM;
    const float sA = inv[M0 + mn];     // fold mean-normalization into A load

    v8f acc = {};

#if HAVE_WMMA_F32X4
    for (int k = 0; k < NDIM; k += 4) {
        v2f a, b;
        // aggregate path:  (agg .* inv) @ Wl
        a.x = arow[k + khalf] * sA;
        a.y = arow[k + khalf + 1] * sA;
        b.x = Wl[(k + khalf) * NDIM + N0 + mn];
        b.y = Wl[(k + khalf + 1) * NDIM + N0 + mn];
        acc = __builtin_amdgcn_wmma_f32_16x16x4_f32(false, a, false, b,
                                                    (short)0, acc, false, false);
        // root path: h @ Wr
        a.x = hrow[k + khalf];
        a.y = hrow[k + khalf + 1];
        b.x = Wr[(k + khalf) * NDIM + N0 + mn];
        b.y = Wr[(k + khalf + 1) * NDIM + N0 + mn];
        acc = __builtin_amdgcn_wmma_f32_16x16x4_f32(false, a, false, b,
                                                    (short)0, acc, false, false);
    }
#else
    // scalar fallback (keeps the same C fragment layout)
    for (int v = 0; v < 8; ++v) {
        const int row = M0 + v + hi;
        const float sr = inv[row];
        float s = 0.f;
        for (int k = 0; k < NDIM; ++k) {
            s += agg[(long)row * NDIM + k] * sr * Wl[k * NDIM + N0 + mn];
            s += h[(long)row * NDIM + k] * Wr[k * NDIM + N0 + mn];
        }
        acc[v] = s;
    }
#endif

    const float bv = bias[N0 + mn];
    for (int v = 0; v < 8; ++v) {
        const int row = M0 + v + hi;
        float r = acc[v] + bv;
        if (do_relu) r = fmaxf(r, 0.f);
        out[(long)row * NDIM + N0 + mn] = r;
    }
}

// ---------------------------------------------------------------------------
// launch
// ---------------------------------------------------------------------------
extern "C" void kernel_launch(void* const* d_in, const int* in_sizes, int n_in,
                              void* d_out, int out_size, void* d_ws, size_t ws_size,
                              hipStream_t stream) {
    const float* x = (const float*)d_in[0];
    const long long* ei = (const long long*)d_in[1];
    const float* Wl1 = (const float*)d_in[2];
    const float* Wr1 = (const float*)d_in[3];
    const float* b1  = (const float*)d_in[4];
    const float* Wl2 = (const float*)d_in[5];
    const float* Wr2 = (const float*)d_in[6];
    const float* b2  = (const float*)d_in[7];

    const int n_nodes = in_sizes[0] / NDIM;      // 20000
    const int n_edges = in_sizes[1] / 2;         // 640000
    const long long* src = ei;
    const long long* dst = ei + n_edges;

    // workspace layout: agg [N*D] | h [N*D] | inv [N]
    float* agg = (float*)d_ws;
    float* h   = agg + (size_t)n_nodes * NDIM;
    float* inv = h + (size_t)n_nodes * NDIM;

    float* outp = (float*)d_out;

    const int feat4 = (n_nodes * NDIM) / 4;
    const int zero_blocks = (feat4 + 255) / 256;

    // 1) degrees -> inverse degrees
    zero_f4<<<(n_nodes / 4 + 255) / 256, 256, 0, stream>>>((float4*)inv, n_nodes / 4);
    degree_kernel<<<2500, 256, 0, stream>>>(dst, inv, n_edges);
    invdeg_kernel<<<(n_nodes + 255) / 256, 256, 0, stream>>>(inv, n_nodes);

    // 2) layer 1: aggregate x, then GEMM + relu -> h
    zero_f4<<<zero_blocks, 256, 0, stream>>>((float4*)agg, feat4);
    scatter_kernel<<<8192, 256, 0, stream>>>(x, src, dst, agg, n_edges);
    const int tiles = (n_nodes / 16) * (NDIM / 16);          // 10000
    sage_layer_kernel<<<tiles / WAVES_PER_BLOCK, WAVES_PER_BLOCK * 32, 0, stream>>>(
        agg, inv, x, Wl1, Wr1, b1, h, 1);

    // 3) layer 2: aggregate h, then GEMM -> out
    zero_f4<<<zero_blocks, 256, 0, stream>>>((float4*)agg, feat4);
    scatter_kernel<<<8192, 256, 0, stream>>>(h, src, dst, agg, n_edges);
    sage_layer_kernel<<<tiles / WAVES_PER_BLOCK, WAVES_PER_BLOCK * 32, 0, stream>>>(
        agg, inv, h, Wl2, Wr2, b2, outp, 0);
}